// RankingLoss_67654324846803
// MI455X (gfx1250) — compile-verified
//
#include <hip/hip_runtime.h>
#include <hip/hip_bf16.h>

#define N_NODES  8192
#define D_FEAT   128
#define N_ANCHOR 512
#define KSEL     32
#define TILE_N   256
#define NTILES   (N_NODES / TILE_N)
#define ROW_STRIDE 129   // 128 dwords + 1 pad dword -> bank = (node + d) % 64, conflict-free

typedef __attribute__((ext_vector_type(2)))  float        v2f;
typedef __attribute__((ext_vector_type(8)))  float        v8f;
typedef __attribute__((ext_vector_type(4)))  unsigned int u32x4;
typedef __attribute__((ext_vector_type(8)))  int          i32x8;
typedef __attribute__((ext_vector_type(4)))  int          i32x4;

#if defined(__has_builtin)
#if __has_builtin(__builtin_amdgcn_tensor_load_to_lds)
#define HAVE_TDM 1
#endif
#endif
#ifndef HAVE_TDM
#define HAVE_TDM 0
#endif

// The *device* pass must have the TDM builtin (host pass only parses stubs).
// Use __HIP_DEVICE_COMPILE__: this toolchain defines __AMDGCN__ on the host
// pass as well, so that macro cannot discriminate the passes.
#if defined(__HIP_DEVICE_COMPILE__) && __HIP_DEVICE_COMPILE__ && !HAVE_TDM
#error "gfx1250 TDM builtin __builtin_amdgcn_tensor_load_to_lds not available"
#endif

// DMA a [TILE_N x D_FEAT] f32 tile, row-major, global -> LDS, inserting one
// pad dword after each 128-dword row (ROW_STRIDE = 129).
__device__ __forceinline__ void tdm_load_tile(const float* gsrc, unsigned lds_byte_off) {
#if HAVE_TDM
    unsigned long long ga = (unsigned long long)(uintptr_t)gsrc;
    u32x4 g0;
    g0.x = 1u;                                                // count=1, user descriptor
    g0.y = lds_byte_off;                                      // lds_addr (bytes)
    g0.z = (unsigned)(ga & 0xFFFFFFFFu);                      // global_addr[31:0]
    g0.w = (unsigned)((ga >> 32) & 0x01FFFFFFu) | (2u << 30); // global_addr[56:32] | type=2
    i32x8 g1;
    // data_size=2 (4B) | pad_enable | pad_interval=6 (128 dwords) | pad_amount=0 (1 dword)
    g1[0] = (int)((2u << 16) | (1u << 20) | (6u << 22));
    g1[1] = (int)(128u << 16);                 // tensor_dim0[15:0]=128 (elems/row)
    g1[2] = (int)(8192u << 16);                // tensor_dim0 hi=0 | tensor_dim1[15:0]=8192
    g1[3] = (int)(128u << 16);                 // tensor_dim1 hi=0 | tile_dim0=128
    g1[4] = (int)(TILE_N);                     // tile_dim1=256 | tile_dim2=0
    g1[5] = (int)(128);                        // tensor_dim0_stride = 128 elems
    g1[6] = 0;                                 // stride hi | tensor_dim1_stride lo
    g1[7] = 0;
    i32x4 g2 = {1, 0, 0, 0};                   // tensor_dim2=1 (unused dims benign)
    i32x4 g3 = {0, 0, 0, 0};
#if __clang_major__ >= 23
    i32x8 g4 = {0, 0, 0, 0, 0, 0, 0, 0};
    __builtin_amdgcn_tensor_load_to_lds(g0, g1, g2, g3, g4, 0);
#else
    __builtin_amdgcn_tensor_load_to_lds(g0, g1, g2, g3, 0);
#endif
#else
    (void)gsrc; (void)lds_byte_off;
#endif
}

__device__ __forceinline__ void tdm_wait_le1() {
#if HAVE_TDM
    __builtin_amdgcn_s_wait_tensorcnt(1);      // oldest outstanding TDM done (in-order)
#endif
}
__device__ __forceinline__ void tdm_wait_0() {
#if HAVE_TDM
    __builtin_amdgcn_s_wait_tensorcnt(0);
#endif
}

// ---------------------------------------------------------------------------
// Kernel 1: D_a = L1(out1[anchor1[a]], out2[anchor2[a]]) + MARGIN, via
// V_WMMA_F32_16X16X4_F32 with an all-ones B (rowsum of abs-diffs).
// One wave handles 16 anchors: A-matrix 16x4 layout -> lane = M | (Kpair<<4).
// ---------------------------------------------------------------------------
__global__ __launch_bounds__(32) void rowdist_wmma_kernel(
    const float* __restrict__ out1, const float* __restrict__ out2,
    const int* __restrict__ anchor1, const int* __restrict__ anchor2,
    float* __restrict__ wsD)
{
    const int lane  = threadIdx.x;        // 0..31, wave32
    const int m     = lane & 15;          // A-matrix row (anchor within group)
    const int kbase = (lane >> 4) * 2;    // lanes 0-15: K=0,1  lanes 16-31: K=2,3
    const int a     = blockIdx.x * 16 + m;
    const float* r1 = out1 + (size_t)anchor1[a] * D_FEAT;
    const float* r2 = out2 + (size_t)anchor2[a] * D_FEAT;

    v8f c = {};
    v2f b = {1.0f, 1.0f};                 // ones -> matmul = row reduction
#pragma unroll 4
    for (int t = 0; t < D_FEAT / 4; ++t) {
        const int d0 = t * 4 + kbase;
        v2f av;
        av.x = fabsf(r1[d0]     - r2[d0]);
        av.y = fabsf(r1[d0 + 1] - r2[d0 + 1]);
        c = __builtin_amdgcn_wmma_f32_16x16x4_f32(
                /*neg_a=*/false, av, /*neg_b=*/false, b,
                /*c_mod=*/(short)0, c, /*reuse_a=*/false, /*reuse_b=*/false);
    }
    // D[m][n] identical across n. C/D layout: VGPR r = rows r (lanes 0-15)
    // and r+8 (lanes 16-31). Pull column 0 of each row and store.
#pragma unroll
    for (int mm = 0; mm < 16; ++mm) {
        float s = __shfl(c[mm & 7], (mm < 8) ? 0 : 16, 32);
        if (lane == mm) wsD[blockIdx.x * 16 + mm] = s + 1.0f;   // + MARGIN
    }
}

// ---------------------------------------------------------------------------
// Kernel 2: one block per (side, anchor). Double-buffered TDM staging of
// 256-node tiles into LDS (padded rows), 8192 L1 distances into LDS, then
// 32 exact arg-min sweeps accumulating relu(D_a - dist).
// ---------------------------------------------------------------------------
__global__ __launch_bounds__(256) void cdist_topk_kernel(
    const float* __restrict__ out1, const float* __restrict__ out2,
    const int* __restrict__ anchor1, const int* __restrict__ anchor2,
    const float* __restrict__ wsD, float* __restrict__ partials)
{
    extern __shared__ float smem[];
    float* anc    = smem;                                        // 128
    float* rv     = smem + 128;                                  // 256
    int*   ri     = (int*)(smem + 384);                          // 256
    float* ytile0 = smem + 640;                                  // 256*129
    float* ytile1 = ytile0 + TILE_N * ROW_STRIDE;                // 256*129
    float* dists  = ytile1 + TILE_N * ROW_STRIDE;                // 8192

    const int tid  = threadIdx.x;
    const int side = blockIdx.x >> 9;                            // 0: a1 vs out2, 1: a2 vs out1
    const int a    = blockIdx.x & (N_ANCHOR - 1);
    const float* Ab = side ? out2 : out1;
    const float* Yb = side ? out1 : out2;
    const int aidx  = side ? anchor2[a] : anchor1[a];

    if (tid < D_FEAT) anc[tid] = Ab[(size_t)aidx * D_FEAT + tid];

    // Prime the pipeline: tile 0 -> buffer 0.
    if (tid == 0) tdm_load_tile(Yb, (unsigned)(uintptr_t)(void*)ytile0);
    __syncthreads();

    for (int t = 0; t < NTILES; ++t) {
        float* buf = (t & 1) ? ytile1 : ytile0;
        if (tid == 0) {
            if (t + 1 < NTILES) {
                // Prefetch next tile into the other buffer (safe: the barrier
                // ending iteration t-1 drained all readers of that buffer),
                // then wait for the *current* tile only -> DMA overlaps compute.
                float* nbuf = (t & 1) ? ytile0 : ytile1;
                tdm_load_tile(Yb + (size_t)(t + 1) * TILE_N * D_FEAT,
                              (unsigned)(uintptr_t)(void*)nbuf);
                tdm_wait_le1();
            } else {
                tdm_wait_0();
            }
        }
        __syncthreads();

        // thread tid owns node t*256+tid; bank = (tid + d) % 64 -> conflict-free
        const float* yr = buf + tid * ROW_STRIDE;
        float acc = 0.0f;
#pragma unroll 8
        for (int d = 0; d < D_FEAT; ++d) acc += fabsf(anc[d] - yr[d]);
        dists[t * TILE_N + tid] = acc;
        __syncthreads();                                         // drain readers of buf
    }

    // Exact selection of the 32 smallest; contribution relu(Dm - dist) each.
    const float Dm = wsD[a];
    float partial = 0.0f;                                        // meaningful on tid 0
    for (int k = 0; k < KSEL; ++k) {
        float mv = 3.402823466e38f;
        int   mi = 0;
        for (int i = tid; i < N_NODES; i += 256) {
            const float v = dists[i];
            if (v < mv) { mv = v; mi = i; }
        }
        rv[tid] = mv; ri[tid] = mi;
        __syncthreads();
        for (int s = 128; s > 0; s >>= 1) {
            if (tid < s && rv[tid + s] < rv[tid]) { rv[tid] = rv[tid + s]; ri[tid] = ri[tid + s]; }
            __syncthreads();
        }
        if (tid == 0) {
            const float m = rv[0];
            dists[ri[0]] = 3.402823466e38f;                      // remove selected instance
            const float contrib = Dm - m;
            partial += contrib > 0.0f ? contrib : 0.0f;
        }
        __syncthreads();
    }
    if (tid == 0) partials[blockIdx.x] = partial;
}

// ---------------------------------------------------------------------------
// Kernel 3: deterministic fixed-order reduction of the 1024 block partials.
// ---------------------------------------------------------------------------
__global__ __launch_bounds__(512) void finalize_kernel(
    const float* __restrict__ partials, float* __restrict__ out)
{
    __shared__ float s[512];
    const int t = threadIdx.x;
    s[t] = partials[t] + partials[t + 512];
    __syncthreads();
    for (int w = 256; w > 0; w >>= 1) {
        if (t < w) s[t] += s[t + w];
        __syncthreads();
    }
    if (t == 0) out[0] = s[0] * (1.0f / (float)(N_ANCHOR * KSEL));
}

extern "C" void kernel_launch(void* const* d_in, const int* in_sizes, int n_in,
                              void* d_out, int out_size, void* d_ws, size_t ws_size,
                              hipStream_t stream) {
    const float* out1    = (const float*)d_in[0];
    const float* out2    = (const float*)d_in[1];
    const int*   anchor1 = (const int*)d_in[2];
    const int*   anchor2 = (const int*)d_in[3];
    float* outp = (float*)d_out;

    float* wsD      = (float*)d_ws;            // 512 floats
    float* partials = (float*)d_ws + N_ANCHOR; // 1024 floats

    // 512 anchors, 16 per wave, 1 wave per block
    rowdist_wmma_kernel<<<N_ANCHOR / 16, 32, 0, stream>>>(out1, out2, anchor1, anchor2, wsD);

    // (anc 128 + rv 256 + ri 256 + 2*256*129 tiles + 8192 dists) floats = 299520 B
    const size_t smem_bytes =
        (size_t)(640 + 2 * TILE_N * ROW_STRIDE + N_NODES) * sizeof(float);
    cdist_topk_kernel<<<2 * N_ANCHOR, 256, smem_bytes, stream>>>(
        out1, out2, anchor1, anchor2, wsD, partials);

    finalize_kernel<<<1, 512, 0, stream>>>(partials, outp);
    (void)in_sizes; (void)n_in; (void)out_size; (void)ws_size;
}